// PCFEALoss_no_mean_46042049413621
// MI455X (gfx1250) — compile-verified
//
#include <hip/hip_runtime.h>
#include <hip/hip_bf16.h>

typedef __attribute__((ext_vector_type(2))) float v2f;
typedef __attribute__((ext_vector_type(8))) float v8f;

#define N_ROWS 4096
#define NCLS   64
#define NFEA   256

// ---------------------------------------------------------------------------
// Kernel A: Saug[y][c] = 0.5 * Lambda * (w_c - w_y)^T CV[y] (w_c - w_y)
// One workgroup per class y (64 blocks, 8 waves). T = D * CV[y] via
// V_WMMA_F32_16X16X4_F32, then row-wise dot with D using the C/D VGPR layout.
// ---------------------------------------------------------------------------
__global__ __launch_bounds__(256) void isda_sigma_kernel(
    const float* __restrict__ W,        // [NCLS, NFEA]
    const float* __restrict__ CV,       // [NCLS, NFEA, NFEA]
    const float* __restrict__ lambda_p, // [1]
    float* __restrict__ Saug)           // [NCLS, NCLS]
{
  const int y    = blockIdx.x;
  const int tid  = threadIdx.x;
  const int w    = tid >> 5;     // wave id 0..7
  const int lane = tid & 31;
  const int half = lane >> 4;    // 0: lanes 0-15, 1: lanes 16-31
  const int lrow = lane & 15;

  __shared__ float Spart[2][NCLS];

  const float* cvy = CV + (size_t)y * NFEA * NFEA;
  const float* wy  = W + y * NFEA;

  // wave -> (M tile, half of the 16 N tiles)
  const int mt  = w & 3;   // M tile 0..3  (rows mt*16 .. mt*16+15)
  const int nth = w >> 2;  // N-tile half: tiles [8*nth, 8*nth+8)

  const int m = mt * 16 + lrow;          // A-fragment row for this lane
  const float* wm = W + m * NFEA;

  float rowsum[8];
#pragma unroll
  for (int r = 0; r < 8; ++r) rowsum[r] = 0.0f;

  for (int nti = 0; nti < 8; ++nti) {
    const int nt = nth * 8 + nti;
    const int n  = nt * 16 + lrow;       // B-fragment / output column
    v8f acc = {};
    for (int k0 = 0; k0 < NFEA; k0 += 4) {
      const int k = k0 + 2 * half;       // lanes 0-15: K=k0,k0+1; 16-31: K=k0+2,k0+3
      v2f a, b;
      a.x = wm[k]     - wy[k];           // A = D (16x4 tile, f32 layout)
      a.y = wm[k + 1] - wy[k + 1];
      b.x = cvy[(size_t)k       * NFEA + n];  // B = CV (4x16 tile)
      b.y = cvy[(size_t)(k + 1) * NFEA + n];
      acc = __builtin_amdgcn_wmma_f32_16x16x4_f32(
          /*neg_a=*/false, a, /*neg_b=*/false, b,
          /*c_mod=*/(short)0, acc, /*reuse_a=*/false, /*reuse_b=*/false);
    }
    // C/D layout: VGPR r, lanes 0-15 -> M=r, lanes 16-31 -> M=r+8; N = lrow.
#pragma unroll
    for (int r = 0; r < 8; ++r) {
      const int crow = mt * 16 + half * 8 + r;
      const float d  = W[crow * NFEA + n] - wy[n];
      rowsum[r] += acc[r] * d;
    }
  }

  // Butterfly-reduce each row partial across its 16-lane group (deterministic).
#pragma unroll
  for (int r = 0; r < 8; ++r) {
#pragma unroll
    for (int off = 1; off < 16; off <<= 1)
      rowsum[r] += __shfl_xor(rowsum[r], off, 32);
  }
  if (lrow == 0) {
#pragma unroll
    for (int r = 0; r < 8; ++r)
      Spart[nth][mt * 16 + half * 8 + r] = rowsum[r];
  }
  __syncthreads();

  if (tid < NCLS) {
    const float lam = lambda_p[0];
    Saug[y * NCLS + tid] = 0.5f * lam * (Spart[0][tid] + Spart[1][tid]);
  }
}

// ---------------------------------------------------------------------------
// Kernel B: per-row augmented cross-entropy.  aug[c] = pred[n,c] + Saug[y_n,c]
// nll[n] = logsumexp(aug) - aug[y_n]
// ---------------------------------------------------------------------------
__global__ __launch_bounds__(256) void isda_ce_rows(
    const float* __restrict__ pred,   // [N_ROWS, NCLS]
    const int*   __restrict__ labels, // [N_ROWS]
    const float* __restrict__ Saug,   // [NCLS, NCLS]
    float* __restrict__ nll)          // [N_ROWS]
{
  const int n = blockIdx.x * blockDim.x + threadIdx.x;
  if (n >= N_ROWS) return;
  const int y = labels[n];
  const float* p = pred + n * NCLS;
  const float* s = Saug + y * NCLS;

  float m = -3.402823466e38f;
  for (int c = 0; c < NCLS; ++c) m = fmaxf(m, p[c] + s[c]);
  float sum = 0.0f;
  for (int c = 0; c < NCLS; ++c) sum += expf(p[c] + s[c] - m);
  nll[n] = m + logf(sum) - (p[y] + s[y]);
}

// ---------------------------------------------------------------------------
// Kernel C: deterministic mean over N_ROWS
// ---------------------------------------------------------------------------
__global__ __launch_bounds__(256) void isda_reduce_mean(
    const float* __restrict__ nll, float* __restrict__ out)
{
  __shared__ float sh[256];
  float s = 0.0f;
  for (int i = threadIdx.x; i < N_ROWS; i += 256) s += nll[i];
  sh[threadIdx.x] = s;
  __syncthreads();
  for (int stride = 128; stride > 0; stride >>= 1) {
    if ((int)threadIdx.x < stride) sh[threadIdx.x] += sh[threadIdx.x + stride];
    __syncthreads();
  }
  if (threadIdx.x == 0) out[0] = sh[0] / (float)N_ROWS;
}

extern "C" void kernel_launch(void* const* d_in, const int* in_sizes, int n_in,
                              void* d_out, int out_size, void* d_ws, size_t ws_size,
                              hipStream_t stream) {
  (void)in_sizes; (void)n_in; (void)out_size; (void)ws_size;
  const float* W        = (const float*)d_in[0]; // fc_weight [64,256]
  // d_in[1] = features (unused by the reference computation)
  const float* pred     = (const float*)d_in[2]; // [4096,64]
  const int*   labels   = (const int*)  d_in[3]; // [4096]
  const float* lambda_p = (const float*)d_in[4]; // [1]
  const float* CV       = (const float*)d_in[5]; // [64,256,256]

  float* Saug = (float*)d_ws;        // 64*64 floats
  float* nll  = Saug + NCLS * NCLS;  // 4096 floats

  isda_sigma_kernel<<<NCLS, 256, 0, stream>>>(W, CV, lambda_p, Saug);
  isda_ce_rows<<<(N_ROWS + 255) / 256, 256, 0, stream>>>(pred, labels, Saug, nll);
  isda_reduce_mean<<<1, 256, 0, stream>>>(nll, (float*)d_out);
}